// NoRefRetIQANet_70849780515286
// MI455X (gfx1250) — compile-verified
//
#include <hip/hip_runtime.h>
#include <stddef.h>
#include <stdint.h>

typedef __attribute__((ext_vector_type(2))) float v2f;
typedef __attribute__((ext_vector_type(8))) float v8f;

#define TOPK 9
#define KC   64      // k-chunk (floats) staged in LDS per iteration
#define LDB  68      // padded LDS row stride in floats -> conflict-free ds_load_b64

// ---------------------------------------------------------------------------
// inverse L2 norm of each row of x [rows, D] -> inv[row] = 1/||x_row||
// ---------------------------------------------------------------------------
__global__ __launch_bounds__(256) void rownorm_inv_kernel(
    const float* __restrict__ x, float* __restrict__ inv, int D)
{
    __shared__ float red[256];
    const int row = blockIdx.x;
    const float* r = x + (size_t)row * (size_t)D;
    float s = 0.0f;
    for (int i = threadIdx.x; i < D; i += 256) {
        float v = r[i];
        s += v * v;
    }
    red[threadIdx.x] = s;
    __syncthreads();
    for (int off = 128; off > 0; off >>= 1) {
        if ((int)threadIdx.x < off) red[threadIdx.x] += red[threadIdx.x + off];
        __syncthreads();
    }
    if (threadIdx.x == 0) inv[row] = rsqrtf(red[0]);
}

// ---------------------------------------------------------------------------
// Cosine-similarity GEMM via V_WMMA_F32_16X16X4_F32 with double-buffered
// async global->LDS staging of the DB tile (ASYNCcnt path, no VGPR roundtrip).
//   S[m, n] = (Q[m,:] . DB[n,:]) * qinv[m] * dbinv[n]
// Block = 128 threads = 4 waves; block owns 16 DB columns; wave w computes
// query rows [16w, 16w+16). DB tile in LDS is reused by all 4 waves.
// Steady-state loop is branchless: last chunk is peeled into an epilogue.
// ---------------------------------------------------------------------------
__global__ __launch_bounds__(128) void cosgemm_wmma_kernel(
    const float* __restrict__ Q,     // [64, D]
    const float* __restrict__ DB,    // [N,  D]
    const float* __restrict__ qinv,  // [64]
    const float* __restrict__ dbinv, // [N]
    float* __restrict__ S,           // [64, N]
    int D, int N)
{
    __shared__ float Bs[2][16 * LDB];

    const int n0   = blockIdx.x * 16;
    const int tid  = threadIdx.x;
    const int wave = tid >> 5;       // 0..3
    const int lane = tid & 31;
    const int m0   = wave * 16;
    const int kh   = lane >> 4;      // K-half select (A/B fragment layout)
    const int l16  = lane & 15;

    // This thread's two float4 slots of the 16 x KC tile:
    //   linear = tid       -> row 0..7,  linear = tid + 128 -> row 8..15
    const int r0 = tid >> 4;
    const int r1 = 8 + (tid >> 4);
    const int cc = (tid & 15) * 4;   // float column within chunk

    const float* g0 = DB + (size_t)(n0 + r0) * (size_t)D + (size_t)cc;
    const float* g1 = DB + (size_t)(n0 + r1) * (size_t)D + (size_t)cc;

    // LDS byte addresses (low 32 bits of generic shared pointer == LDS offset)
    unsigned lds0[2], lds1[2];
    lds0[0] = (unsigned)(uintptr_t)&Bs[0][r0 * LDB + cc];
    lds1[0] = (unsigned)(uintptr_t)&Bs[0][r1 * LDB + cc];
    lds0[1] = (unsigned)(uintptr_t)&Bs[1][r0 * LDB + cc];
    lds1[1] = (unsigned)(uintptr_t)&Bs[1][r1 * LDB + cc];

    // A-fragment base: lane holds Q[m0+l16][k + 2*kh .. +1]
    const float* qrow = Q + (size_t)(m0 + l16) * (size_t)D + (size_t)(kh * 2);

    const int nch = D / KC;          // >= 2 for all our shapes (D = 2048/4096)

#define ISSUE_ASYNC(buf, koff)                                                 \
    do {                                                                       \
        asm volatile("global_load_async_to_lds_b128 %0, %1, off"               \
                     :: "v"(lds0[(buf)]), "v"(g0 + (koff)) : "memory");        \
        asm volatile("global_load_async_to_lds_b128 %0, %1, off"               \
                     :: "v"(lds1[(buf)]), "v"(g1 + (koff)) : "memory");        \
    } while (0)

#define COMPUTE_CHUNK(buf, cidx)                                               \
    do {                                                                       \
        const float* bs = &Bs[(buf)][0];                                       \
        const float* qk = qrow + (cidx) * KC;                                  \
        _Pragma("unroll")                                                      \
        for (int k = 0; k < KC; k += 4) {                                      \
            v2f a = *(const v2f*)(qk + k);                                     \
            v2f b;                                                             \
            b.x = bs[l16 * LDB + k + kh * 2];                                  \
            b.y = bs[l16 * LDB + k + kh * 2 + 1];                              \
            acc = __builtin_amdgcn_wmma_f32_16x16x4_f32(                       \
                false, a, false, b, (short)0, acc, false, false);              \
        }                                                                      \
    } while (0)

    v8f acc = {};

    ISSUE_ASYNC(0, 0);

    // steady state: chunks 0 .. nch-2 (unconditional wait/issue, branchless body)
    for (int c = 0; c < nch - 1; ++c) {
        const int cur = c & 1;
        ISSUE_ASYNC(cur ^ 1, (c + 1) * KC);
        // chunk c's 2 loads done; the 2 just-issued may stay outstanding
        asm volatile("s_wait_asynccnt 0x2" ::: "memory");
        __syncthreads();   // everyone's async writes to Bs[cur] now visible

        // L2 prefetch two chunks ahead (gfx1250 global_prefetch_b8); clamped
        const int pf = (c + 2 < nch) ? (c + 2) : (nch - 1);
        __builtin_prefetch(g0 + (size_t)pf * KC, 0, 1);
        __builtin_prefetch(g1 + (size_t)pf * KC, 0, 1);

        COMPUTE_CHUNK(cur, c);
        __syncthreads();   // all waves done with Bs[cur] before it is refilled
    }

    // epilogue: last chunk
    {
        const int c = nch - 1;
        asm volatile("s_wait_asynccnt 0x0" ::: "memory");
        __syncthreads();
        COMPUTE_CHUNK(c & 1, c);
    }
#undef ISSUE_ASYNC
#undef COMPUTE_CHUNK

    // C/D layout: VGPR v -> M = v + 8*kh, N = l16 (lanes 0-15 / 16-31)
    const float dbs = dbinv[n0 + l16];
#pragma unroll
    for (int v = 0; v < 8; ++v) {
        int m = m0 + v + kh * 8;
        S[(size_t)m * (size_t)N + (size_t)(n0 + l16)] = acc[v] * qinv[m] * dbs;
    }
}

// ---------------------------------------------------------------------------
// Top-9 (descending, lower index wins ties) of one row of S [., N]
// ---------------------------------------------------------------------------
__global__ __launch_bounds__(256) void topk9_kernel(
    const float* __restrict__ S, int N, int* __restrict__ idxOut)
{
    const int row = blockIdx.x;
    const float* s = S + (size_t)row * (size_t)N;

    float v[TOPK];
    int   id[TOPK];
#pragma unroll
    for (int i = 0; i < TOPK; ++i) { v[i] = -__builtin_inff(); id[i] = 0x7fffffff; }

    for (int n = threadIdx.x; n < N; n += 256) {
        float x = s[n];
        if (x > v[TOPK - 1]) {
            int j = TOPK - 1;
            while (j > 0 && v[j - 1] < x) { v[j] = v[j - 1]; id[j] = id[j - 1]; --j; }
            v[j] = x; id[j] = n;
        }
    }

    __shared__ float sv[256];
    __shared__ int   si[256];
    __shared__ int   so[256];

    int p = 0;
    for (int r = 0; r < TOPK; ++r) {
        sv[threadIdx.x] = (p < TOPK) ? v[p]  : -__builtin_inff();
        si[threadIdx.x] = (p < TOPK) ? id[p] : 0x7fffffff;
        so[threadIdx.x] = threadIdx.x;
        __syncthreads();
        for (int off = 128; off > 0; off >>= 1) {
            if ((int)threadIdx.x < off) {
                float av = sv[threadIdx.x], bv = sv[threadIdx.x + off];
                bool take = (bv > av) ||
                            (bv == av && si[threadIdx.x + off] < si[threadIdx.x]);
                if (take) {
                    sv[threadIdx.x] = bv;
                    si[threadIdx.x] = si[threadIdx.x + off];
                    so[threadIdx.x] = so[threadIdx.x + off];
                }
            }
            __syncthreads();
        }
        if (threadIdx.x == 0) idxOut[row * TOPK + r] = si[0];
        if ((int)threadIdx.x == so[0]) ++p;   // winner pops its candidate
        __syncthreads();
    }
}

// ---------------------------------------------------------------------------
// Gather metrics, interleave sem/dst, mean.  d_out = [result(64) | retrieved(64x18)]
// ---------------------------------------------------------------------------
__global__ __launch_bounds__(64) void finalize_kernel(
    const int* __restrict__ idx_sem, const int* __restrict__ idx_dst,
    const float* __restrict__ metrics,
    float* __restrict__ result, float* __restrict__ retrieved)
{
    const int b = threadIdx.x;   // 64 threads
    float sum = 0.0f;
#pragma unroll
    for (int j = 0; j < TOPK; ++j) {
        float ms = metrics[idx_sem[b * TOPK + j]];
        float md = metrics[idx_dst[b * TOPK + j]];
        retrieved[b * 2 * TOPK + 2 * j]     = ms;
        retrieved[b * 2 * TOPK + 2 * j + 1] = md;
        sum += ms + md;
    }
    result[b] = sum * (1.0f / (2.0f * TOPK));
}

// ---------------------------------------------------------------------------
extern "C" void kernel_launch(void* const* d_in, const int* in_sizes, int n_in,
                              void* d_out, int out_size, void* d_ws, size_t ws_size,
                              hipStream_t stream)
{
    const float* f_content     = (const float*)d_in[0]; // [64, Ds]
    const float* f_distorsion  = (const float*)d_in[1]; // [64, Dd]
    const float* semantic_db   = (const float*)d_in[2]; // [N, Ds]
    const float* distorsion_db = (const float*)d_in[3]; // [N, Dd]
    const float* metrics       = (const float*)d_in[4]; // [N]

    const int B  = 64;
    const int Ds = in_sizes[0] / B;
    const int Dd = in_sizes[1] / B;
    const int N  = in_sizes[4];

    // workspace layout
    float* qsi  = (float*)d_ws;            // [B]   query sem inv-norm
    float* qdi  = qsi + B;                 // [B]   query dst inv-norm
    float* dsi  = qdi + B;                 // [N]   sem db inv-norm
    float* ddi  = dsi + N;                 // [N]   dst db inv-norm
    float* Ssem = ddi + N;                 // [B*N] sem cosine scores
    float* Sdst = Ssem + (size_t)B * N;    // [B*N] dst cosine scores
    int*   isem = (int*)(Sdst + (size_t)B * N); // [B*TOPK]
    int*   idst = isem + B * TOPK;              // [B*TOPK]

    float* result    = (float*)d_out;      // [B]
    float* retrieved = result + B;         // [B, 2*TOPK]

    // 1) inverse row norms
    rownorm_inv_kernel<<<B, 256, 0, stream>>>(f_content,     qsi, Ds);
    rownorm_inv_kernel<<<B, 256, 0, stream>>>(f_distorsion,  qdi, Dd);
    rownorm_inv_kernel<<<N, 256, 0, stream>>>(semantic_db,   dsi, Ds);
    rownorm_inv_kernel<<<N, 256, 0, stream>>>(distorsion_db, ddi, Dd);

    // 2) cosine-similarity GEMMs (fp32 WMMA, async LDS staging)
    cosgemm_wmma_kernel<<<N / 16, 128, 0, stream>>>(
        f_content, semantic_db, qsi, dsi, Ssem, Ds, N);
    cosgemm_wmma_kernel<<<N / 16, 128, 0, stream>>>(
        f_distorsion, distorsion_db, qdi, ddi, Sdst, Dd, N);

    // 3) per-row top-9
    topk9_kernel<<<B, 256, 0, stream>>>(Ssem, N, isem);
    topk9_kernel<<<B, 256, 0, stream>>>(Sdst, N, idst);

    // 4) gather + mean
    finalize_kernel<<<1, 64, 0, stream>>>(isem, idst, metrics, result, retrieved);
}